// LayerPropertiesScattering_53360673685856
// MI455X (gfx1250) — compile-verified
//
#include <hip/hip_runtime.h>
#include <hip/hip_bf16.h>

#define NCH  29
#define EPSF 1e-7f
#define TPB  128           // 4 waves per block
#define WPB  4
#define SPW  64            // samples per wave (two 32-sample pairs)

typedef float v2f __attribute__((ext_vector_type(2)));
typedef float v8f __attribute__((ext_vector_type(8)));

// D = A(16x4 f32) * B(4x16 f32) + C(16x16 f32)
static __device__ __forceinline__ v8f wmma4(v2f a, v2f b, v8f c) {
  return __builtin_amdgcn_wmma_f32_16x16x4_f32(false, a, false, b, (short)0, c, false, false);
}
// half-wave swap: lane <-> lane^16 (ds_swizzle SWAPX16, immediate, no index VGPR)
static __device__ __forceinline__ float sx16(float v) {
  return __int_as_float(__builtin_amdgcn_ds_swizzle(__float_as_int(v), 0x401F));
}
static __device__ __forceinline__ float fast_rcp(float v) {
  return __builtin_amdgcn_rcpf(v);
}
static __device__ __forceinline__ void cpy(float* dst, const float* __restrict__ src, int n) {
  for (int i = threadIdx.x; i < n; i += TPB) dst[i] = src[i];
}

// float offsets of one MLP's weight block in LDS
#define OW0 0
#define OB0 1015
#define OW1 1160
#define OB1 1740
#define OW2 1856
#define OB2 2320
#define OW3 2436
#define OB3 2784
#define MLPSZ 2871

__global__ __launch_bounds__(TPB)
void lps_kernel(const float* __restrict__ g_tau,
                const float* __restrict__ g_mu,
                const float* __restrict__ g_mub,
                const float* __restrict__ g_lw,  const float* __restrict__ g_h2o,
                const float* __restrict__ g_o3,  const float* __restrict__ g_co2,
                const float* __restrict__ g_u,   const float* __restrict__ g_n2o,
                const float* __restrict__ g_ch4,
                const float* __restrict__ Wd0, const float* __restrict__ bd0,
                const float* __restrict__ Wf0, const float* __restrict__ bf0,
                const float* __restrict__ Wd1, const float* __restrict__ bd1,
                const float* __restrict__ Wf1, const float* __restrict__ bf1,
                const float* __restrict__ Wd2, const float* __restrict__ bd2,
                const float* __restrict__ Wf2, const float* __restrict__ bf2,
                const float* __restrict__ Wd3, const float* __restrict__ bd3,
                const float* __restrict__ Wf3, const float* __restrict__ bf3,
                float* __restrict__ out)
{
  __shared__ float sW[2 * MLPSZ];              // weights+biases, both MLPs (~23 KB)
  __shared__ float sTau[WPB * SPW * NCH];      // per-wave 64-sample tau tile (~29 KB)

  cpy(sW + 0*MLPSZ + OW0, Wd0, 29*7*5);  cpy(sW + 0*MLPSZ + OB0, bd0, 29*5);
  cpy(sW + 0*MLPSZ + OW1, Wd1, 29*5*4);  cpy(sW + 0*MLPSZ + OB1, bd1, 29*4);
  cpy(sW + 0*MLPSZ + OW2, Wd2, 29*4*4);  cpy(sW + 0*MLPSZ + OB2, bd2, 29*4);
  cpy(sW + 0*MLPSZ + OW3, Wd3, 29*4*3);  cpy(sW + 0*MLPSZ + OB3, bd3, 29*3);
  cpy(sW + 1*MLPSZ + OW0, Wf0, 29*7*5);  cpy(sW + 1*MLPSZ + OB0, bf0, 29*5);
  cpy(sW + 1*MLPSZ + OW1, Wf1, 29*5*4);  cpy(sW + 1*MLPSZ + OB1, bf1, 29*4);
  cpy(sW + 1*MLPSZ + OW2, Wf2, 29*4*4);  cpy(sW + 1*MLPSZ + OB2, bf2, 29*4);
  cpy(sW + 1*MLPSZ + OW3, Wf3, 29*4*3);  cpy(sW + 1*MLPSZ + OB3, bf3, 29*3);

  const int lane = threadIdx.x & 31;
  const int wave = threadIdx.x >> 5;
  const int nb   = blockIdx.x * (WPB * SPW) + wave * SPW;  // wave's first sample
  const int n0   = nb + lane;                              // pair-0 sample of this lane

  // coalesced tau staging (64 samples x 29 channels)
  float* stau = sTau + wave * (SPW * NCH);
  for (int i = lane; i < SPW * NCH; i += 32)
    stau[i] = g_tau[(size_t)nb * NCH + i];

  __syncthreads();

  const bool  lo   = (lane < 16);
  const int   M    = lane & 15;
  const float hone = lo ? 1.0f : 0.0f;
  const v2f   bc1  = { hone, 0.0f };   // bias-row B operand for layers 2/3 (K=4 is ones-row)

  float invmu[2], invmub[2];
  v2f B0[2][2][2], B1[2][2][2];        // [pair][mlp][chain] layer-0 B ops (X^T with ones-row K=7)
  #pragma unroll
  for (int s = 0; s < 2; ++s) {
    const int n = n0 + 32 * s;
    invmu[s]  = 1.0f / (g_mu[n]  + EPSF);
    invmub[s] = 1.0f / (g_mub[n] + EPSF);
    float cc[7] = { g_lw[n], g_h2o[n], g_o3[n], g_co2[n], g_u[n], g_n2o[n], g_ch4[n] };
    #pragma unroll
    for (int m = 0; m < 2; ++m) {
      const float sc = m ? invmub[s] : invmu[s];
      float x[7], t[7];
      #pragma unroll
      for (int f = 0; f < 7; ++f) { x[f] = cc[f] * sc; t[f] = sx16(x[f]); }
      v2f bA0 = { lo ? x[0] : t[2],  lo ? x[1] : t[3] };
      v2f bA1 = { lo ? x[4] : t[6],  lo ? x[5] : 1.0f };   // K=7 is ones (bias) row
      v2f bB0 = { lo ? t[0] : x[2],  lo ? t[1] : x[3] };
      v2f bB1 = { lo ? t[4] : x[6],  lo ? t[5] : 1.0f };
      B0[s][m][0] = bA0; B1[s][m][0] = bA1;
      B0[s][m][1] = bB0; B1[s][m][1] = bB1;
    }
  }

  const v8f z = {0.f,0.f,0.f,0.f,0.f,0.f,0.f,0.f};

  #pragma unroll 1
  for (int c = 0; c < NCH; ++c) {
    float tdir[2], tdif[2];
    #pragma unroll
    for (int s = 0; s < 2; ++s) {
      const float tv = stau[(lane + 32 * s) * NCH + c];
      tdir[s] = __expf(-tv * invmu[s]);
      tdif[s] = __expf(-tv * invmub[s]);
    }

    float sm[2][2][3];
    #pragma unroll
    for (int m = 0; m < 2; ++m) {
      const float* WB = sW + m * MLPSZ;
      const float* W0 = WB + OW0; const float* b0 = WB + OB0;
      const float* W1 = WB + OW1; const float* b1 = WB + OB1;
      const float* W2 = WB + OW2; const float* b2 = WB + OB2;
      const float* W3 = WB + OW3; const float* b3 = WB + OB3;

      // ---- per-channel A registers, biases folded as extra K rows ----
      v2f a00={0.f,0.f}, a01={0.f,0.f}, a10={0.f,0.f}, a11={0.f,0.f};
      v2f a20={0.f,0.f}, a21={0.f,0.f}, a30={0.f,0.f}, a31={0.f,0.f};
      if (M < 5) {                       // layer0: W0^T(5x7)+bias col -> K=8
        const float* w = W0 + c*35 + M;  // W0[c][k][m] = w[k*5]
        a00[0] = lo ? w[0]  : w[10];
        a00[1] = lo ? w[5]  : w[15];
        a01[0] = lo ? w[20] : w[30];
        a01[1] = lo ? w[25] : b0[c*5 + M];
      }
      if (M < 4) {                       // layer1: W1^T(4x5)+bias col -> K=6
        const float* w = W1 + c*20 + M;  // w[k*4]
        a10[0] = lo ? w[0]  : w[8];
        a10[1] = lo ? w[4]  : w[12];
        a11[0] = lo ? w[16] : 0.0f;
        a11[1] = lo ? b1[c*4 + M] : 0.0f;
        const float* v = W2 + c*16 + M;  // layer2: W2^T(4x4)+bias col -> K=5
        a20[0] = lo ? v[0]  : v[8];
        a20[1] = lo ? v[4]  : v[12];
        a21[0] = lo ? b2[c*4 + M] : 0.0f;
      }
      if (M < 3) {                       // layer3: W3^T(3x4)+bias col -> K=5
        const float* w = W3 + c*12 + M;  // w[k*3]
        a30[0] = lo ? w[0] : w[6];
        a30[1] = lo ? w[3] : w[9];
        a31[0] = lo ? b3[c*3 + M] : 0.0f;
      }

      #pragma unroll
      for (int s = 0; s < 2; ++s) {
        // ---- layer 0 ----
        v8f dA = wmma4(a01, B1[s][m][0], wmma4(a00, B0[s][m][0], z));
        v8f dB = wmma4(a01, B1[s][m][1], wmma4(a00, B0[s][m][1], z));
        #pragma unroll
        for (int r = 0; r < 5; ++r) { dA[r] = fmaxf(dA[r], 0.f); dB[r] = fmaxf(dB[r], 0.f); }

        // ---- layer 1 (K rows: h0..h4, ones) ----
        float sA2 = sx16(dA[2]), sA3 = sx16(dA[3]);
        float sB2 = sx16(dB[2]), sB3 = sx16(dB[3]);
        v2f c0A = { lo ? dA[0] : sA2,  lo ? dA[1] : sA3 };
        v2f c1A = { lo ? dA[4] : 0.f,  hone };
        v2f c0B = { lo ? dB[0] : sB2,  lo ? dB[1] : sB3 };
        v2f c1B = { lo ? dB[4] : 0.f,  hone };
        dA = wmma4(a11, c1A, wmma4(a10, c0A, z));
        dB = wmma4(a11, c1B, wmma4(a10, c0B, z));
        #pragma unroll
        for (int r = 0; r < 4; ++r) { dA[r] = fmaxf(dA[r], 0.f); dB[r] = fmaxf(dB[r], 0.f); }

        // ---- layer 2 (K rows: h0..h3, ones) ----
        sA2 = sx16(dA[2]); sA3 = sx16(dA[3]);
        sB2 = sx16(dB[2]); sB3 = sx16(dB[3]);
        v2f d0A = { lo ? dA[0] : sA2,  lo ? dA[1] : sA3 };
        v2f d0B = { lo ? dB[0] : sB2,  lo ? dB[1] : sB3 };
        dA = wmma4(a21, bc1, wmma4(a20, d0A, z));
        dB = wmma4(a21, bc1, wmma4(a20, d0B, z));
        #pragma unroll
        for (int r = 0; r < 4; ++r) { dA[r] = fmaxf(dA[r], 0.f); dB[r] = fmaxf(dB[r], 0.f); }

        // ---- layer 3: logits (bias folded) ----
        sA2 = sx16(dA[2]); sA3 = sx16(dA[3]);
        sB2 = sx16(dB[2]); sB3 = sx16(dB[3]);
        v2f e0A = { lo ? dA[0] : sA2,  lo ? dA[1] : sA3 };
        v2f e0B = { lo ? dB[0] : sB2,  lo ? dB[1] : sB3 };
        dA = wmma4(a31, bc1, wmma4(a30, e0A, z));
        dB = wmma4(a31, bc1, wmma4(a30, e0B, z));

        // combine chains (each lane -> its own sample), softmax with fast rcp
        float l0s = sx16(dB[0]), l1s = sx16(dB[1]), l2s = sx16(dB[2]);
        const float l0 = lo ? dA[0] : l0s;
        const float l1 = lo ? dA[1] : l1s;
        const float l2 = lo ? dA[2] : l2s;
        const float mx = fmaxf(l0, fmaxf(l1, l2));
        const float p0 = __expf(l0 - mx), p1 = __expf(l1 - mx), p2 = __expf(l2 - mx);
        const float inv = fast_rcp(p0 + p1 + p2);
        sm[s][m][0] = p0 * inv; sm[s][m][1] = p1 * inv; sm[s][m][2] = p2 * inv;
      }
    }

    // out[n][c][0..7] = {t_direct, t_diffuse, e_direct[0..2], e_diffuse[0..2]}
    #pragma unroll
    for (int s = 0; s < 2; ++s) {
      const size_t off = ((size_t)(n0 + 32 * s) * NCH + c) * 8;
      float4 o0 = make_float4(tdir[s], tdif[s], sm[s][0][0], sm[s][0][1]);
      float4 o1 = make_float4(sm[s][0][2], sm[s][1][0], sm[s][1][1], sm[s][1][2]);
      *reinterpret_cast<float4*>(out + off)     = o0;
      *reinterpret_cast<float4*>(out + off + 4) = o1;
    }
  }
}

extern "C" void kernel_launch(void* const* d_in, const int* in_sizes, int n_in,
                              void* d_out, int out_size, void* d_ws, size_t ws_size,
                              hipStream_t stream) {
  const float* tau = (const float*)d_in[0];
  const float* mu  = (const float*)d_in[1];
  const float* mub = (const float*)d_in[2];
  const float* lw  = (const float*)d_in[3];
  const float* h2o = (const float*)d_in[4];
  const float* o3  = (const float*)d_in[5];
  const float* co2 = (const float*)d_in[6];
  const float* u   = (const float*)d_in[7];
  const float* n2o = (const float*)d_in[8];
  const float* ch4 = (const float*)d_in[9];
  // setup_inputs dict order: Wd0, bd0, Wf0, bf0, Wd1, bd1, Wf1, bf1, ...
  const float* Wd0 = (const float*)d_in[10]; const float* bd0 = (const float*)d_in[11];
  const float* Wf0 = (const float*)d_in[12]; const float* bf0 = (const float*)d_in[13];
  const float* Wd1 = (const float*)d_in[14]; const float* bd1 = (const float*)d_in[15];
  const float* Wf1 = (const float*)d_in[16]; const float* bf1 = (const float*)d_in[17];
  const float* Wd2 = (const float*)d_in[18]; const float* bd2 = (const float*)d_in[19];
  const float* Wf2 = (const float*)d_in[20]; const float* bf2 = (const float*)d_in[21];
  const float* Wd3 = (const float*)d_in[22]; const float* bd3 = (const float*)d_in[23];
  const float* Wf3 = (const float*)d_in[24]; const float* bf3 = (const float*)d_in[25];
  float* out = (float*)d_out;

  const int N = in_sizes[1];                 // 262144 samples
  const int blocks = N / (WPB * SPW);        // 256 samples per block -> 1024 blocks, exact
  lps_kernel<<<blocks, TPB, 0, stream>>>(tau, mu, mub, lw, h2o, o3, co2, u, n2o, ch4,
      Wd0, bd0, Wf0, bf0, Wd1, bd1, Wf1, bf1,
      Wd2, bd2, Wf2, bf2, Wd3, bd3, Wf3, bf3, out);
}